// Model_39676907885408
// MI455X (gfx1250) — compile-verified
//
#include <hip/hip_runtime.h>
#include <hip/hip_bf16.h>
#include <math.h>

typedef float v2f __attribute__((ext_vector_type(2)));
typedef float v8f __attribute__((ext_vector_type(8)));
typedef int   v4i __attribute__((ext_vector_type(4)));

#define GLOBAL_AS __attribute__((address_space(1)))
#define LDS_AS    __attribute__((address_space(3)))

#define SEQ   512
#define BATCH 256                    // 8*8*4 heads
#define BM    128
#define BN    128
#define BK    16
#define LDP   20                     // pad: conflict-free, 8B/16B aligned rows
#define LDBP  132                    // pad for [K][N] tile
#define NKIT  (SEQ / BK)             // 32 K iterations

#define ATTN_OFF ((size_t)BATCH * SEQ * SEQ)       // 67108864
#define MASK_OFF ((size_t)2 * BATCH * SEQ * SEQ)   // 134217728
#define MASK_N   ((size_t)16 * SEQ * SEQ)          // 4194304

static __device__ __forceinline__ v8f zero8() {
    v8f z = {0.f, 0.f, 0.f, 0.f, 0.f, 0.f, 0.f, 0.f};
    return z;
}

static __device__ __forceinline__ void async_cp16(const float* g, float* l) {
    // 16-byte global -> LDS DMA, tracked by ASYNCcnt (no VGPR staging).
    __builtin_amdgcn_global_load_async_to_lds_b128(
        (GLOBAL_AS v4i*)g, (LDS_AS v4i*)l, 0, 0);
}

static __device__ __forceinline__ void wait_async0() {
    asm volatile("s_wait_asynccnt 0x0" ::: "memory");
}

// ---------------- kernel 0: constant attn_mask output ----------------
__global__ __launch_bounds__(256) void fill_mask_kernel(float* __restrict__ mask) {
    size_t i = (size_t)blockIdx.x * blockDim.x + threadIdx.x;
    float4 v = make_float4(-10000.f, -10000.f, -10000.f, -10000.f);
    ((float4*)mask)[i] = v;
}

// ---------------- kernel 1: scores = (Q @ Q^T) * 1/sqrt(S) ----------------
// One 128x128 C tile per 256-thread block; 8 waves, each owns 32(M) x 64(N).
// Double-buffered LDS tiles filled by async global->LDS DMA.
__global__ __launch_bounds__(256) void scores_nt_kernel(const float* __restrict__ q,
                                                        float* __restrict__ w) {
    __shared__ float As[2][BM * LDP];
    __shared__ float Bs[2][BN * LDP];

    const int batch = blockIdx.y;
    const int tm = blockIdx.x >> 2;
    const int tn = blockIdx.x & 3;
    const float* Q = q + (size_t)batch * SEQ * SEQ;
    float* W = w + (size_t)batch * SEQ * SEQ;
    const int m0 = tm * BM, n0 = tn * BN;

    const int tid  = threadIdx.x;
    const int wid  = tid >> 5;
    const int lane = tid & 31;
    const int wm   = wid & 3;        // 4 waves along M (32 rows each)
    const int wn   = wid >> 2;       // 2 waves along N (64 cols each)
    const int half = lane >> 4;
    const int ln   = lane & 15;

    v8f acc[2][4];
#pragma unroll
    for (int i = 0; i < 2; ++i)
#pragma unroll
        for (int j = 0; j < 4; ++j) acc[i][j] = zero8();

    const int lrow = tid >> 2;           // 0..63
    const int lcg  = (tid & 3) * 4;      // 0,4,8,12

    // prologue: fill buffer 0
#pragma unroll
    for (int r = 0; r < 2; ++r) {
        int row = lrow + r * 64;
        async_cp16(Q + (size_t)(m0 + row) * SEQ + lcg, &As[0][row * LDP + lcg]);
        async_cp16(Q + (size_t)(n0 + row) * SEQ + lcg, &Bs[0][row * LDP + lcg]);
    }
    wait_async0();
    __syncthreads();

    for (int it = 0; it < NKIT; ++it) {
        const int buf = it & 1;
        if (it + 1 < NKIT) {
            const int k1 = (it + 1) * BK;
#pragma unroll
            for (int r = 0; r < 2; ++r) {
                int row = lrow + r * 64;
                async_cp16(Q + (size_t)(m0 + row) * SEQ + k1 + lcg,
                           &As[buf ^ 1][row * LDP + lcg]);
                async_cp16(Q + (size_t)(n0 + row) * SEQ + k1 + lcg,
                           &Bs[buf ^ 1][row * LDP + lcg]);
            }
        }
#pragma unroll
        for (int kk = 0; kk < BK; kk += 4) {
            v2f afr[2], bfr[4];
#pragma unroll
            for (int i = 0; i < 2; ++i)
                afr[i] = *(const v2f*)(&As[buf][(wm * 32 + i * 16 + ln) * LDP + kk + half * 2]);
#pragma unroll
            for (int j = 0; j < 4; ++j)
                bfr[j] = *(const v2f*)(&Bs[buf][(wn * 64 + j * 16 + ln) * LDP + kk + half * 2]);
#pragma unroll
            for (int i = 0; i < 2; ++i)
#pragma unroll
                for (int j = 0; j < 4; ++j)
                    acc[i][j] = __builtin_amdgcn_wmma_f32_16x16x4_f32(
                        false, afr[i], false, bfr[j], (short)0, acc[i][j], false, false);
        }
        wait_async0();
        __syncthreads();
    }

    const float scale = 0.04419417382415922f;   // 1/sqrt(512)
#pragma unroll
    for (int i = 0; i < 2; ++i)
#pragma unroll
        for (int j = 0; j < 4; ++j)
#pragma unroll
            for (int v = 0; v < 8; ++v) {
                int row = m0 + wm * 32 + i * 16 + half * 8 + v;
                int col = n0 + wn * 64 + j * 16 + ln;
                W[(size_t)row * SEQ + col] = acc[i][j][v] * scale;
            }
}

// ---------------- kernel 2: row softmax in place (wave per row) ----------------
__global__ __launch_bounds__(256) void softmax_rows_kernel(float* __restrict__ w) {
    const int wid  = threadIdx.x >> 5;
    const int lane = threadIdx.x & 31;
    const size_t row = (size_t)blockIdx.x * 8 + wid;
    float* p = w + row * SEQ;

    float4 v[4];
    float m = -INFINITY;
#pragma unroll
    for (int i = 0; i < 4; ++i) {
        v[i] = ((const float4*)p)[lane + i * 32];
        m = fmaxf(m, fmaxf(fmaxf(v[i].x, v[i].y), fmaxf(v[i].z, v[i].w)));
    }
#pragma unroll
    for (int off = 16; off > 0; off >>= 1) m = fmaxf(m, __shfl_xor(m, off, 32));

    float s = 0.f;
#pragma unroll
    for (int i = 0; i < 4; ++i) {
        v[i].x = __expf(v[i].x - m);
        v[i].y = __expf(v[i].y - m);
        v[i].z = __expf(v[i].z - m);
        v[i].w = __expf(v[i].w - m);
        s += v[i].x + v[i].y + v[i].z + v[i].w;
    }
#pragma unroll
    for (int off = 16; off > 0; off >>= 1) s += __shfl_xor(s, off, 32);

    const float inv = 1.0f / s;
#pragma unroll
    for (int i = 0; i < 4; ++i) {
        v[i].x *= inv; v[i].y *= inv; v[i].z *= inv; v[i].w *= inv;
        ((float4*)p)[lane + i * 32] = v[i];
    }
}

// ---------------- kernel 3: output = P @ Q (NN GEMM) ----------------
__global__ __launch_bounds__(256) void av_nn_kernel(const float* __restrict__ w,
                                                    const float* __restrict__ q,
                                                    float* __restrict__ o) {
    __shared__ float As[2][BM * LDP];
    __shared__ float Bs[2][BK * LDBP];

    const int batch = blockIdx.y;
    const int tm = blockIdx.x >> 2;
    const int tn = blockIdx.x & 3;
    const float* W = w + (size_t)batch * SEQ * SEQ;
    const float* Q = q + (size_t)batch * SEQ * SEQ;
    float* O = o + (size_t)batch * SEQ * SEQ;
    const int m0 = tm * BM, n0 = tn * BN;

    const int tid  = threadIdx.x;
    const int wid  = tid >> 5;
    const int lane = tid & 31;
    const int wm   = wid & 3;
    const int wn   = wid >> 2;
    const int half = lane >> 4;
    const int ln   = lane & 15;

    v8f acc[2][4];
#pragma unroll
    for (int i = 0; i < 2; ++i)
#pragma unroll
        for (int j = 0; j < 4; ++j) acc[i][j] = zero8();

    const int larow = tid >> 2;          // A loader: 0..63
    const int lacg  = (tid & 3) * 4;
    const int lbrow = tid >> 5;          // B loader: 0..7
    const int lbcg  = (tid & 31) * 4;    // 0..124

    // prologue: fill buffer 0
#pragma unroll
    for (int r = 0; r < 2; ++r) {
        int arow = larow + r * 64;
        async_cp16(W + (size_t)(m0 + arow) * SEQ + lacg, &As[0][arow * LDP + lacg]);
        int brow = lbrow + r * 8;
        async_cp16(Q + (size_t)brow * SEQ + n0 + lbcg, &Bs[0][brow * LDBP + lbcg]);
    }
    wait_async0();
    __syncthreads();

    for (int it = 0; it < NKIT; ++it) {
        const int buf = it & 1;
        if (it + 1 < NKIT) {
            const int k1 = (it + 1) * BK;
#pragma unroll
            for (int r = 0; r < 2; ++r) {
                int arow = larow + r * 64;
                async_cp16(W + (size_t)(m0 + arow) * SEQ + k1 + lacg,
                           &As[buf ^ 1][arow * LDP + lacg]);
                int brow = lbrow + r * 8;
                async_cp16(Q + (size_t)(k1 + brow) * SEQ + n0 + lbcg,
                           &Bs[buf ^ 1][brow * LDBP + lbcg]);
            }
        }
#pragma unroll
        for (int kk = 0; kk < BK; kk += 4) {
            v2f afr[2], bfr[4];
#pragma unroll
            for (int i = 0; i < 2; ++i)
                afr[i] = *(const v2f*)(&As[buf][(wm * 32 + i * 16 + ln) * LDP + kk + half * 2]);
            const int kb = kk + half * 2;
#pragma unroll
            for (int j = 0; j < 4; ++j) {
                int nidx = wn * 64 + j * 16 + ln;
                v2f b;
                b.x = Bs[buf][kb * LDBP + nidx];
                b.y = Bs[buf][(kb + 1) * LDBP + nidx];
                bfr[j] = b;
            }
#pragma unroll
            for (int i = 0; i < 2; ++i)
#pragma unroll
                for (int j = 0; j < 4; ++j)
                    acc[i][j] = __builtin_amdgcn_wmma_f32_16x16x4_f32(
                        false, afr[i], false, bfr[j], (short)0, acc[i][j], false, false);
        }
        wait_async0();
        __syncthreads();
    }

#pragma unroll
    for (int i = 0; i < 2; ++i)
#pragma unroll
        for (int j = 0; j < 4; ++j)
#pragma unroll
            for (int v = 0; v < 8; ++v) {
                int row = m0 + wm * 32 + i * 16 + half * 8 + v;
                int col = n0 + wn * 64 + j * 16 + ln;
                O[(size_t)row * SEQ + col] = acc[i][j][v];
            }
}

extern "C" void kernel_launch(void* const* d_in, const int* in_sizes, int n_in,
                              void* d_out, int out_size, void* d_ws, size_t ws_size,
                              hipStream_t stream) {
    (void)in_sizes; (void)n_in; (void)out_size; (void)d_ws; (void)ws_size;
    const float* q = (const float*)d_in[1];      // d_in[0] = l1 (unused by outputs)
    float* out    = (float*)d_out;               // output region
    float* attn   = out + ATTN_OFF;              // attn_weight region
    float* mask   = out + MASK_OFF;              // attn_mask region

    fill_mask_kernel<<<(int)(MASK_N / 4 / 256), 256, 0, stream>>>(mask);

    dim3 grid(16, BATCH);                         // 4x4 tiles of 128, 256 heads
    scores_nt_kernel<<<grid, 256, 0, stream>>>(q, attn);

    softmax_rows_kernel<<<(int)(BATCH * SEQ / 8), 256, 0, stream>>>(attn);

    av_nn_kernel<<<grid, 256, 0, stream>>>(attn, q, out);
}